// SyllableComposer_74586402063028
// MI455X (gfx1250) — compile-verified
//
#include <hip/hip_runtime.h>
#include <hip/hip_bf16.h>

namespace {

constexpr int Bz   = 16;
constexpr int Sq   = 2048;
constexpr int Dd   = 512;
constexpr int NSg  = Sq + 1;            // 2049 segments per batch (incl. segment 0)
constexpr int NROW = Bz * NSg;          // 32784 segment rows
constexpr int NT64 = (NROW + 63) / 64;  // 513 macro-tiles of 64 rows
// NROW % 64 == 16: the single partial macro-tile has exactly sub-tile 0 valid.

typedef float v2f __attribute__((ext_vector_type(2)));
typedef float v8f __attribute__((ext_vector_type(8)));

// ---------------------------------------------------------------------------
// Kernel 0: workspace init (counts=0, smin=INT_MAX, smax=-1, S buffers = 0)
// ---------------------------------------------------------------------------
__global__ void __launch_bounds__(256)
init_kernel(unsigned* __restrict__ counts, int* __restrict__ smin,
            int* __restrict__ smax, float4* __restrict__ zero4, size_t n4)
{
    const size_t tid    = (size_t)blockIdx.x * blockDim.x + threadIdx.x;
    const size_t stride = (size_t)gridDim.x * blockDim.x;
    const float4 z = make_float4(0.f, 0.f, 0.f, 0.f);
    for (size_t i = tid; i < n4; i += stride) zero4[i] = z;
    for (size_t i = tid; i < (size_t)NROW; i += stride) {
        counts[i] = 0u;
        smin[i]   = 0x7fffffff;
        smax[i]   = -1;
    }
}

// ---------------------------------------------------------------------------
// Kernel 1: per-batch segment scan. One wave32 per batch; ballot-based
// inclusive prefix sum of "start" flags over S=2048.
// ---------------------------------------------------------------------------
__global__ void __launch_bounds__(32)
scan_kernel(const float* __restrict__ pos, int* __restrict__ seg,
            unsigned* __restrict__ counts, int* __restrict__ smin,
            int* __restrict__ smax)
{
    const int b    = blockIdx.x;
    const int lane = threadIdx.x;
    int base = 0;
    for (int s0 = 0; s0 < Sq; s0 += 32) {
        const int s = s0 + lane;
        const size_t pi = ((size_t)b * Sq + s) * 3;
        const float x0 = pos[pi + 0], x1 = pos[pi + 1], x2 = pos[pi + 2];
        const float mx = fmaxf(x0, fmaxf(x1, x2));
        const float e0 = expf(x0 - mx), e1 = expf(x1 - mx), e2 = expf(x2 - mx);
        const float p0 = e0 / (e0 + e1 + e2);
        const int start = (p0 > 0.5f) ? 1 : 0;
        const unsigned long long bal = __ballot(start);
        const int pre  = __popcll(bal & ((1ull << lane) - 1ull));
        const int incl = base + pre + start;
        const int valid = ((x0 + x1 + x2) != 0.0f) ? 1 : 0;
        const int sg = valid ? incl : 0;
        seg[(size_t)b * Sq + s] = sg;
        const int idx = b * NSg + sg;
        atomicAdd(&counts[idx], 1u);
        atomicMin(&smin[idx], s);
        atomicMax(&smax[idx], s);
        base += __popcll(bal);
    }
}

// ---------------------------------------------------------------------------
// Kernel 2: segment-sum of probability-scaled features.
// Block = (s-chunk of 128, d-chunk of 128, batch). Register accumulate per
// current segment; atomic flush on segment change.
// ---------------------------------------------------------------------------
__global__ void __launch_bounds__(128)
accum_kernel(const float* __restrict__ feat, const float* __restrict__ pos,
             const int* __restrict__ seg,
             float* __restrict__ S0, float* __restrict__ S1, float* __restrict__ S2)
{
    const int b     = blockIdx.z;
    const int dbase = blockIdx.y * 128;
    const int sbeg  = blockIdx.x * 128;
    const int d     = dbase + threadIdx.x;

    float ai = 0.f, am = 0.f, af = 0.f;
    int cur = -1;
    for (int s = sbeg; s < sbeg + 128; ++s) {
        const size_t rowi = (size_t)b * Sq + s;
        const int sg = seg[rowi];
        const float x0 = pos[rowi * 3 + 0], x1 = pos[rowi * 3 + 1], x2 = pos[rowi * 3 + 2];
        const float mx = fmaxf(x0, fmaxf(x1, x2));
        const float e0 = expf(x0 - mx), e1 = expf(x1 - mx), e2 = expf(x2 - mx);
        const float inv = 1.f / (e0 + e1 + e2);
        const float f = feat[rowi * Dd + d];
        if (sg != cur) {
            if (cur >= 0) {
                const size_t o = ((size_t)(b * NSg + cur)) * Dd + d;
                atomicAdd(&S0[o], ai);
                atomicAdd(&S1[o], am);
                atomicAdd(&S2[o], af);
            }
            cur = sg;
            ai = am = af = 0.f;
        }
        ai += f * e0 * inv;
        am += f * e1 * inv;
        af += f * e2 * inv;
    }
    if (cur >= 0) {
        const size_t o = ((size_t)(b * NSg + cur)) * Dd + d;
        atomicAdd(&S0[o], ai);
        atomicAdd(&S1[o], am);
        atomicAdd(&S2[o], af);
    }
}

// ---------------------------------------------------------------------------
// Kernel 3: fp32 WMMA GEMM, Y(+)= A(NROWx512) @ W(512x512, row stride 512).
// 64-row macro-tile per wave (4 sub-tiles of 16): one B fragment feeds four
// v_wmma_f32_16x16x4_f32 ops. A macro-tile staged to LDS with the CDNA5
// async DMA path (GLOBAL_LOAD_ASYNC_TO_LDS_B128 + S_WAIT_ASYNCCNT).
// Tile index is readfirstlane'd so the full/partial epilogue is a scalar
// branch and store bases live in SGPRs (no EXEC predication).
// WITH_BIAS: add counts[row]*bias[col]. ACCUM: Y += result.
// ---------------------------------------------------------------------------
template <bool WITH_BIAS, bool ACCUM>
__global__ void __launch_bounds__(64)
gemm64_kernel(const float* A, const float* __restrict__ W,
              const float* __restrict__ bias,       // used iff WITH_BIAS
              const unsigned* __restrict__ counts,  // used iff WITH_BIAS
              float* Y)
{
    __shared__ float As[2 * 64 * Dd];  // 256 KB (<= 320 KB WGP budget)
    // Wave id is physically uniform; readfirstlane makes it uniform for the
    // compiler too -> scalar branches + SGPR store bases downstream.
    const int wave = __builtin_amdgcn_readfirstlane((int)(threadIdx.x >> 5));
    const int lane = threadIdx.x & 31;
    const int t64  = blockIdx.x * 2 + wave;
    if (t64 >= NT64) return;
    const int  r0    = t64 * 64;
    const bool full  = (r0 + 64 <= NROW);
    const int  nrows = full ? 64 : (NROW - r0);   // 64 or 16
    float* sA = As + (size_t)wave * 64 * Dd;

    // --- Stage A macro-tile: async global->LDS DMA, 16B per lane per issue.
    // Trip count (nrows*128/32) is wave-uniform, so EXEC stays all-ones.
    {
        const unsigned lbase = (unsigned)(uintptr_t)sA;  // low 32 bits = LDS byte addr
        const float4* gs = (const float4*)(A + (size_t)r0 * Dd);
        const int n16 = nrows * (Dd / 4);
        for (int t = lane; t < n16; t += 32) {
            const unsigned laddr = lbase + (unsigned)t * 16u;
            asm volatile("global_load_async_to_lds_b128 %0, %1, off"
                         :: "v"(laddr), "v"(gs + t)
                         : "memory");
        }
        asm volatile("s_wait_asynccnt 0x0" ::: "memory");
    }

    const int half = lane >> 4;   // K-half select (K+{0,1} vs K+{2,3})
    const int lm   = lane & 15;   // M / N index within 16

    // Preload bias scale factors (segment counts) once per tile.
    float cnts[4][8];
    if (WITH_BIAS) {
        if (full) {
#pragma unroll
            for (int t = 0; t < 4; ++t)
#pragma unroll
                for (int j = 0; j < 8; ++j)
                    cnts[t][j] = (float)counts[r0 + t * 16 + j + 8 * half];
        } else {
            // Partial tile: only sub-tile 0 is consumed; its 16 rows are valid.
#pragma unroll
            for (int j = 0; j < 8; ++j) cnts[0][j] = (float)counts[r0 + j + 8 * half];
#pragma unroll
            for (int t = 1; t < 4; ++t)
#pragma unroll
                for (int j = 0; j < 8; ++j) cnts[t][j] = 0.f;
        }
    }

    for (int nc = 0; nc < Dd / 16; ++nc) {
        const int col = nc * 16 + lm;
        v8f acc0 = {0.f, 0.f, 0.f, 0.f, 0.f, 0.f, 0.f, 0.f};
        v8f acc1 = acc0, acc2 = acc0, acc3 = acc0;
        const float* wp = W + (size_t)(2 * half) * Dd + col;
        const float* ap = sA + (size_t)lm * Dd + 2 * half;
#pragma unroll 4
        for (int kk = 0; kk < Dd / 4; ++kk) {
            v2f bv;
            bv.x = wp[0];
            bv.y = wp[Dd];
            const v2f a0 = *(const v2f*)(ap + 0 * 16 * Dd);
            const v2f a1 = *(const v2f*)(ap + 1 * 16 * Dd);
            const v2f a2 = *(const v2f*)(ap + 2 * 16 * Dd);
            const v2f a3 = *(const v2f*)(ap + 3 * 16 * Dd);
            acc0 = __builtin_amdgcn_wmma_f32_16x16x4_f32(false, a0, false, bv, (short)0, acc0, false, false);
            acc1 = __builtin_amdgcn_wmma_f32_16x16x4_f32(false, a1, false, bv, (short)0, acc1, false, false);
            acc2 = __builtin_amdgcn_wmma_f32_16x16x4_f32(false, a2, false, bv, (short)0, acc2, false, false);
            acc3 = __builtin_amdgcn_wmma_f32_16x16x4_f32(false, a3, false, bv, (short)0, acc3, false, false);
            wp += 4 * Dd;
            ap += 4;
        }
        const float bcol = WITH_BIAS ? bias[col] : 0.f;
        v8f accs[4] = {acc0, acc1, acc2, acc3};
        if (full) {
            // Scalar-branch fast path, no per-lane predication.
#pragma unroll
            for (int t = 0; t < 4; ++t)
#pragma unroll
                for (int j = 0; j < 8; ++j) {
                    const int row = r0 + t * 16 + j + 8 * half;
                    float v = accs[t][j];
                    if (WITH_BIAS) v += cnts[t][j] * bcol;
                    float* yp = Y + (size_t)row * Dd + col;
                    if (ACCUM) v += *yp;
                    *yp = v;
                }
        } else {
            // Partial macro-tile: exactly sub-tile 0 is valid (NROW%64==16).
#pragma unroll
            for (int j = 0; j < 8; ++j) {
                const int row = r0 + j + 8 * half;
                float v = accs[0][j];
                if (WITH_BIAS) v += cnts[0][j] * bcol;
                float* yp = Y + (size_t)row * Dd + col;
                if (ACCUM) v += *yp;
                *yp = v;
            }
        }
    }
}

// ---------------------------------------------------------------------------
// Kernel 4: per-segment-row gates. gates = softmax([Y0,Y1,Y2] @ Wg + bg),
// then scale each channel row in place. One wave per row.
// ---------------------------------------------------------------------------
__global__ void __launch_bounds__(256)
gates_kernel(float* Y0, float* Y1, float* Y2,
             const float* __restrict__ Wg, const float* __restrict__ bg)
{
    const int wave = threadIdx.x >> 5;
    const int lane = threadIdx.x & 31;
    const int r = blockIdx.x * 8 + wave;

    float t0 = 0.f, t1 = 0.f, t2 = 0.f;
    float* Ys[3] = {Y0, Y1, Y2};
    for (int c = 0; c < 3; ++c) {
        const float* y = Ys[c] + (size_t)r * Dd;
        for (int k = lane; k < Dd; k += 32) {
            const float v = y[k];
            const float* w = Wg + (size_t)(c * Dd + k) * 3;
            t0 += v * w[0];
            t1 += v * w[1];
            t2 += v * w[2];
        }
    }
    for (int off = 16; off > 0; off >>= 1) {
        t0 += __shfl_xor(t0, off, 32);
        t1 += __shfl_xor(t1, off, 32);
        t2 += __shfl_xor(t2, off, 32);
    }
    t0 += bg[0]; t1 += bg[1]; t2 += bg[2];
    const float mx = fmaxf(t0, fmaxf(t1, t2));
    const float e0 = expf(t0 - mx), e1 = expf(t1 - mx), e2 = expf(t2 - mx);
    const float inv = 1.f / (e0 + e1 + e2);
    const float g[3] = {e0 * inv, e1 * inv, e2 * inv};
    for (int c = 0; c < 3; ++c) {
        float* y = Ys[c] + (size_t)r * Dd;
        for (int k = lane; k < Dd; k += 32) y[k] *= g[c];
    }
}

// ---------------------------------------------------------------------------
// Kernel 5: output epilogue. Drop segment 0, add bc, zero empty segments.
// ---------------------------------------------------------------------------
__global__ void __launch_bounds__(256)
outepi_kernel(const float* __restrict__ Z, const float* __restrict__ bc,
              const unsigned* __restrict__ counts, float* __restrict__ out)
{
    const size_t i = (size_t)blockIdx.x * 256 + threadIdx.x;  // over B*S*D
    const int d = (int)(i & (Dd - 1));
    const size_t rowOut = i >> 9;               // b*2048 + (n-1)
    const int b = (int)(rowOut >> 11);
    const int n = (int)(rowOut & 2047) + 1;
    const int row = b * NSg + n;
    out[i] = (counts[row] > 0) ? (Z[(size_t)row * Dd + d] + bc[d]) : 0.f;
}

// ---------------------------------------------------------------------------
// Kernel 6: boundary output: [seg_min, seg_max+1] or -1 for empty segments.
// ---------------------------------------------------------------------------
__global__ void __launch_bounds__(256)
bnd_kernel(const unsigned* __restrict__ counts, const int* __restrict__ smin,
           const int* __restrict__ smax, float* __restrict__ bnd)
{
    const int i = blockIdx.x * 256 + threadIdx.x;  // over B*S
    const int b = i >> 11;
    const int n = (i & 2047) + 1;
    const int idx = b * NSg + n;
    const bool ne = counts[idx] > 0;
    bnd[2 * i + 0] = ne ? (float)smin[idx] : -1.f;
    bnd[2 * i + 1] = ne ? (float)(smax[idx] + 1) : -1.f;
}

}  // namespace

extern "C" void kernel_launch(void* const* d_in, const int* in_sizes, int n_in,
                              void* d_out, int out_size, void* d_ws, size_t ws_size,
                              hipStream_t stream)
{
    const float* feat = (const float*)d_in[0];
    const float* pos  = (const float*)d_in[1];
    const float* Wi   = (const float*)d_in[2];
    const float* bi   = (const float*)d_in[3];
    const float* Wm   = (const float*)d_in[4];
    const float* bm   = (const float*)d_in[5];
    const float* Wf   = (const float*)d_in[6];
    const float* bf   = (const float*)d_in[7];
    const float* Wg   = (const float*)d_in[8];
    const float* bg   = (const float*)d_in[9];
    const float* Wc   = (const float*)d_in[10];
    const float* bc   = (const float*)d_in[11];

    char* ws = (char*)d_ws;
    auto alloc = [&](size_t bytes) -> char* {
        char* p = ws;
        ws += (bytes + 255) & ~(size_t)255;
        return p;
    };
    int*      seg    = (int*)alloc((size_t)Bz * Sq * sizeof(int));
    unsigned* counts = (unsigned*)alloc((size_t)NROW * sizeof(unsigned));
    int*      smin   = (int*)alloc((size_t)NROW * sizeof(int));
    int*      smax   = (int*)alloc((size_t)NROW * sizeof(int));
    const size_t sbytes = (size_t)NROW * Dd * sizeof(float);  // 256-aligned
    float* S0 = (float*)alloc(sbytes);
    float* S1 = (float*)alloc(sbytes);
    float* S2 = (float*)alloc(sbytes);
    float* Z  = (float*)alloc(sbytes);

    float* outFeat = (float*)d_out;
    float* outBnd  = outFeat + (size_t)Bz * Sq * Dd;

    // 0) init workspace (S0..S2 contiguous -> single zeroing span)
    init_kernel<<<4096, 256, 0, stream>>>(counts, smin, smax, (float4*)S0,
                                          (size_t)3 * NROW * Dd / 4);
    // 1) segment scan
    scan_kernel<<<Bz, 32, 0, stream>>>(pos, seg, counts, smin, smax);
    // 2) segment-sum of scaled features
    accum_kernel<<<dim3(Sq / 128, Dd / 128, Bz), 128, 0, stream>>>(
        feat, pos, seg, S0, S1, S2);
    // 3) GEMM stage 1 (in place): sum_c = segsum_c @ W_c + counts * b_c
    const int gblocks = (NT64 + 1) / 2;  // 257
    gemm64_kernel<true, false><<<gblocks, 64, 0, stream>>>(S0, Wi, bi, counts, S0);
    gemm64_kernel<true, false><<<gblocks, 64, 0, stream>>>(S1, Wm, bm, counts, S1);
    gemm64_kernel<true, false><<<gblocks, 64, 0, stream>>>(S2, Wf, bf, counts, S2);
    // 4) gates + in-place scaling
    gates_kernel<<<NROW / 8, 256, 0, stream>>>(S0, S1, S2, Wg, bg);
    // 5) GEMM stage 2: Z = sum_c gated_c @ Wc[c*512:(c+1)*512, :]
    gemm64_kernel<false, false><<<gblocks, 64, 0, stream>>>(S0, Wc + 0 * Dd * Dd, nullptr, nullptr, Z);
    gemm64_kernel<false, true ><<<gblocks, 64, 0, stream>>>(S1, Wc + 1 * Dd * Dd, nullptr, nullptr, Z);
    gemm64_kernel<false, true ><<<gblocks, 64, 0, stream>>>(S2, Wc + 2 * Dd * Dd, nullptr, nullptr, Z);
    // 6) epilogue: +bc, mask empty, drop segment 0 -> syllable_features
    outepi_kernel<<<(int)(((size_t)Bz * Sq * Dd) / 256), 256, 0, stream>>>(
        Z, bc, counts, outFeat);
    // 7) boundaries
    bnd_kernel<<<(Bz * Sq) / 256, 256, 0, stream>>>(counts, smin, smax, outBnd);
}